// RelGraphConvBasis_86938728005790
// MI455X (gfx1250) — compile-verified
//
#include <hip/hip_runtime.h>

#define N_NODES 50000
#define N_EDGES 1600000
#define DIM_D   128
#define DIM_F   128
#define N_REL   32
#define N_BAS   8
#define DFSZ    (DIM_D * DIM_F)
#define EPB     128   // edges per block = 8 tiles of 16 (bins aligned to this)
#define LDA     132   // LDS row stride (floats) -> conflict-free column reads

typedef __attribute__((ext_vector_type(2))) float v2f;
typedef __attribute__((ext_vector_type(8))) float v8f;

static __device__ __forceinline__ v8f wmma_f32(v2f a, v2f b, v8f c) {
  // D = A(16x4 f32) * B(4x16 f32) + C(16x16 f32)
  return __builtin_amdgcn_wmma_f32_16x16x4_f32(false, a, false, b, (short)0, c,
                                               false, false);
}

// ---------------- Kernel 1: W_rel[r] = sum_b w_comp[r,b] * weight[b] --------
__global__ __launch_bounds__(256) void compose_wrel(
    const float* __restrict__ wcomp, const float* __restrict__ weight,
    float* __restrict__ wrel) {
  int idx = blockIdx.x * 256 + threadIdx.x;  // R*DF = 524288 total
  int r = idx / DFSZ;
  int df = idx - r * DFSZ;
  float s = 0.f;
#pragma unroll
  for (int b = 0; b < N_BAS; ++b)
    s += wcomp[r * N_BAS + b] * weight[b * DFSZ + df];
  wrel[idx] = s;
}

// ---------------- Kernel 2: histogram over relations ------------------------
__global__ __launch_bounds__(256) void hist_kernel(
    const int* __restrict__ etype, int* __restrict__ cnt) {
  int e = blockIdx.x * 256 + threadIdx.x;
  if (e < N_EDGES) atomicAdd(&cnt[etype[e]], 1);
}

// ---------------- Kernel 3: EPB-aligned exclusive prefix sums ---------------
__global__ void offsets_kernel(const int* __restrict__ cnt,
                               int* __restrict__ off) {
  if (threadIdx.x == 0) {
    int a = 0;
    off[0] = 0;
    for (int r = 0; r < N_REL; ++r) {
      int c = (cnt[r] + EPB - 1) / EPB * EPB;  // pad each bin to 128 edges
      a += c;
      off[r + 1] = a;
    }
  }
}

// ---------------- Kernel 4: bin edge ids by relation ------------------------
__global__ __launch_bounds__(256) void scatter_kernel(
    const int* __restrict__ etype, const int* __restrict__ off,
    int* __restrict__ cur, int* __restrict__ binned) {
  int e = blockIdx.x * 256 + threadIdx.x;
  if (e < N_EDGES) {
    int r = etype[e];
    int pos = off[r] + atomicAdd(&cur[r], 1);
    binned[pos] = e;
  }
}

// ---------------- Kernel 5: per-relation edge-tile GEMM (WMMA f32) ----------
// Block = 128 binned edges of one relation. 8 waves; wave w owns columns
// [16w,16w+16) of W_rel[r], preloaded as 32 B-fragments in VGPRs and reused
// across 8 A-tiles. A tile (16 edges x 128, scaled by norm) staged in LDS.
// 16x128 results atomically accumulated into L2-resident acc[N,F].
__global__ __launch_bounds__(256) void edge_gemm(
    const float* __restrict__ h, const int* __restrict__ src,
    const int* __restrict__ dst, const float* __restrict__ norm,
    const float* __restrict__ wrel, const int* __restrict__ off,
    const int* __restrict__ binned, float* __restrict__ acc) {
  __shared__ float As[16][LDA];
  __shared__ int sdst[16];
  __shared__ int srel;

  const int tid = threadIdx.x;
  const int lane = tid & 31;
  const int wave = tid >> 5;
  const int base = blockIdx.x * EPB;

  if (tid == 0) {
    int r = N_REL - 1;
    for (int i = 0; i < N_REL; ++i) {
      if (base < off[i + 1]) { r = i; break; }
    }
    srel = r;
  }
  __syncthreads();
  const int rel = srel;

  // Preload this wave's B slab: 16 cols x 128 K as 32 v2f fragments.
  const float* Wr = wrel + rel * DFSZ;          // [K=128][N=128] row-major
  const int colbase = wave * 16;
  const int n = colbase + (lane & 15);
  const int khalf = (lane < 16) ? 0 : 2;        // ISA 32-bit A/B lane split
  v2f bfrag[32];
#pragma unroll
  for (int t = 0; t < 32; ++t) {
    int kk = 4 * t + khalf;
    bfrag[t].x = Wr[kk * DIM_F + n];
    bfrag[t].y = Wr[(kk + 1) * DIM_F + n];
  }

  const int arow = tid >> 4;                    // 0..15
  const int ac0 = (tid & 15) * 8;               // 8 floats per thread
  const int am = lane & 15;

  for (int tile = 0; tile < 8; ++tile) {
    const int tbase = base + tile * 16;
    // ---- cooperative A load: 16 edges x 128, scaled by norm ----
    const int e = binned[tbase + arow];
    if ((tid & 15) == 0) sdst[arow] = (e >= 0) ? dst[e] : -1;
    if (e >= 0) {
      const float nm = norm[e];
      const float* hp = h + (long)src[e] * DIM_D + ac0;
      const float4 p0 = *(const float4*)(hp);
      const float4 p1 = *(const float4*)(hp + 4);
      As[arow][ac0 + 0] = p0.x * nm;
      As[arow][ac0 + 1] = p0.y * nm;
      As[arow][ac0 + 2] = p0.z * nm;
      As[arow][ac0 + 3] = p0.w * nm;
      As[arow][ac0 + 4] = p1.x * nm;
      As[arow][ac0 + 5] = p1.y * nm;
      As[arow][ac0 + 6] = p1.z * nm;
      As[arow][ac0 + 7] = p1.w * nm;
    } else {
#pragma unroll
      for (int j = 0; j < 8; ++j) As[arow][ac0 + j] = 0.f;
    }
    __syncthreads();

    // ---- 16x16 output tile per wave: 32 x wmma_f32_16x16x4 ----
    v8f cacc = {0.f, 0.f, 0.f, 0.f, 0.f, 0.f, 0.f, 0.f};
#pragma unroll
    for (int t = 0; t < 32; ++t) {
      v2f af;
      const int kk = 4 * t + khalf;
      af.x = As[am][kk];
      af.y = As[am][kk + 1];
      cacc = wmma_f32(af, bfrag[t], cacc);
    }

    // ---- scatter-accumulate into acc[dst, :] (L2-resident) ----
#pragma unroll
    for (int i = 0; i < 8; ++i) {
      const int m = (lane < 16) ? i : (8 + i);   // C/D VGPR layout
      const int dr = sdst[m];
      if (dr >= 0) unsafeAtomicAdd(&acc[(long)dr * DIM_F + n], cacc[i]);
    }
    __syncthreads();
  }
}

// ---------------- Kernel 6: self-loop GEMM + bias + acc + ReLU --------------
__global__ __launch_bounds__(256) void selfloop_epilogue(
    const float* __restrict__ h, const float* __restrict__ loopw,
    const float* __restrict__ bias, const float* __restrict__ acc,
    float* __restrict__ outp) {
  __shared__ float As[16][LDA];

  const int tid = threadIdx.x;
  const int lane = tid & 31;
  const int wave = tid >> 5;
  const int rowbase = blockIdx.x * 16;

  const int arow = tid >> 4;
  const int ac0 = (tid & 15) * 8;
  const int grow = rowbase + arow;
  if (grow < N_NODES) {
    const float* hp = h + (long)grow * DIM_D + ac0;
    const float4 p0 = *(const float4*)(hp);
    const float4 p1 = *(const float4*)(hp + 4);
    As[arow][ac0 + 0] = p0.x;
    As[arow][ac0 + 1] = p0.y;
    As[arow][ac0 + 2] = p0.z;
    As[arow][ac0 + 3] = p0.w;
    As[arow][ac0 + 4] = p1.x;
    As[arow][ac0 + 5] = p1.y;
    As[arow][ac0 + 6] = p1.z;
    As[arow][ac0 + 7] = p1.w;
  } else {
#pragma unroll
    for (int j = 0; j < 8; ++j) As[arow][ac0 + j] = 0.f;
  }
  __syncthreads();

  const int colbase = wave * 16;
  const int n = colbase + (lane & 15);
  const int khalf = (lane < 16) ? 0 : 2;
  const int am = lane & 15;

  v8f cacc = {0.f, 0.f, 0.f, 0.f, 0.f, 0.f, 0.f, 0.f};
#pragma unroll
  for (int t = 0; t < 32; ++t) {
    const int kk = 4 * t + khalf;
    v2f af, bf;
    af.x = As[am][kk];
    af.y = As[am][kk + 1];
    bf.x = loopw[kk * DIM_F + n];        // loop_weight L2-hot
    bf.y = loopw[(kk + 1) * DIM_F + n];
    cacc = wmma_f32(af, bf, cacc);
  }

  const float bn = bias[n];
#pragma unroll
  for (int i = 0; i < 8; ++i) {
    const int m = (lane < 16) ? i : (8 + i);
    const int gr = rowbase + m;
    if (gr < N_NODES) {
      float v = cacc[i] + bn + acc[(long)gr * DIM_F + n];
      outp[(long)gr * DIM_F + n] = v > 0.f ? v : 0.f;
    }
  }
}

// ---------------- Host launcher ---------------------------------------------
extern "C" void kernel_launch(void* const* d_in, const int* in_sizes, int n_in,
                              void* d_out, int out_size, void* d_ws,
                              size_t ws_size, hipStream_t stream) {
  const float* h      = (const float*)d_in[0];
  const int*   src    = (const int*)d_in[1];
  const int*   dst    = (const int*)d_in[2];
  const int*   etype  = (const int*)d_in[3];
  const float* norm   = (const float*)d_in[4];
  const float* weight = (const float*)d_in[5];
  const float* w_comp = (const float*)d_in[6];
  const float* loopw  = (const float*)d_in[7];
  const float* hbias  = (const float*)d_in[8];
  float* outp = (float*)d_out;
  char* ws = (char*)d_ws;

  // workspace layout (bytes, 256-aligned): total ~34.1 MB
  float* wrel   = (float*)(ws + 0);          //  2,097,152  W_rel [R,D,F]
  float* acc    = (float*)(ws + 2097152);    // 25,600,000  acc   [N,F]
  int*   cnt    = (int*)(ws + 27697152);     //        256
  int*   cur    = (int*)(ws + 27697408);     //        256
  int*   off    = (int*)(ws + 27697664);     //        256
  int*   binned = (int*)(ws + 27697920);     //  6,416,384  (E + R*EPB ids)

  const int cap = N_EDGES + N_REL * EPB;     // 1,604,096 (multiple of EPB)

  hipMemsetAsync(acc, 0, (size_t)N_NODES * DIM_F * sizeof(float), stream);
  hipMemsetAsync(cnt, 0, 768, stream);                       // cnt+cur+off
  hipMemsetAsync(binned, 0xFF, (size_t)cap * sizeof(int), stream);  // -1

  compose_wrel<<<(N_REL * DFSZ) / 256, 256, 0, stream>>>(w_comp, weight, wrel);
  hist_kernel<<<(N_EDGES + 255) / 256, 256, 0, stream>>>(etype, cnt);
  offsets_kernel<<<1, 32, 0, stream>>>(cnt, off);
  scatter_kernel<<<(N_EDGES + 255) / 256, 256, 0, stream>>>(etype, off, cur,
                                                            binned);
  edge_gemm<<<cap / EPB, 256, 0, stream>>>(h, src, dst, norm, wrel, off,
                                           binned, acc);
  selfloop_epilogue<<<(N_NODES + 15) / 16, 256, 0, stream>>>(h, loopw, hbias,
                                                             acc, outp);
  (void)in_sizes; (void)n_in; (void)out_size; (void)ws_size;
}